// CrossModalAttention_2156073582693
// MI455X (gfx1250) — compile-verified
//
#include <hip/hip_runtime.h>
#include <stdint.h>
#include <stddef.h>

// CrossModalAttention for MI455X (gfx1250, wave32).
// All matrix math on the CDNA5 WMMA pipe: V_WMMA_F32_16X16X4_F32.
// BitLinear weights are ternary -> quantized once (exact), scale applied in epilogue.
// Softmax paths kept in f32 (logit sigma ~290; bf16 would mis-blend near-tie rows).
//
// K-slot permutation trick: a WMMA dot product is invariant under permutation of
// logical K as long as A and B fragments use the SAME permutation. Each lane loads
// one contiguous float4 at (k + 4*half); WMMA#1 consumes elements {0,1}, WMMA#2
// elements {2,3}. One b128 load feeds two WMMAs on both operands.

#define B_ 16
#define S_ 512
#define H_ 768
#define NH_ 12
#define HD_ 64
#define QKVW (3 * H_)

typedef float v2f __attribute__((ext_vector_type(2)));
typedef float v4f __attribute__((ext_vector_type(4)));
typedef float v8f __attribute__((ext_vector_type(8)));

static __device__ __forceinline__ v8f wmma4(v2f a, v2f b, v8f c) {
  // (neg_a, A, neg_b, B, c_mod, C, reuse_a, reuse_b)
  return __builtin_amdgcn_wmma_f32_16x16x4_f32(false, a, false, b, (short)0, c,
                                               false, false);
}

// ---------------------------------------------------------------- weight prep
struct WArgs {
  const float* w[8];
  int n[8];
};

__global__ __launch_bounds__(256) void absmean_kernel(WArgs args, float* scales) {
  __shared__ float red[256];
  const int i = blockIdx.x;
  const float* w = args.w[i];
  const int n = args.n[i];
  float acc = 0.f;
  for (int j = threadIdx.x; j < n; j += 256) acc += fabsf(w[j]);
  red[threadIdx.x] = acc;
  __syncthreads();
  for (int o = 128; o > 0; o >>= 1) {
    if (threadIdx.x < o) red[threadIdx.x] += red[threadIdx.x + o];
    __syncthreads();
  }
  if (threadIdx.x == 0) {
    float s = red[0] / (float)n;
    scales[i] = fminf(fmaxf(s, 1e-5f), 1000.f);
  }
}

__global__ __launch_bounds__(256) void quantize_kernel(
    const float* __restrict__ w, float* __restrict__ q,
    const float* __restrict__ sp, int n) {
  const float s = *sp;
  const float thr = 2.0f / 3.0f;
  for (int i = blockIdx.x * 256 + threadIdx.x; i < n; i += gridDim.x * 256) {
    float wn = fminf(fmaxf(w[i] / s, -10.f), 10.f);
    q[i] = (wn > thr ? 1.f : 0.f) - (wn < -thr ? 1.f : 0.f);
  }
}

// --------------------------------------------------------------- WMMA f32 GEMM
// C[MxN] = scale * (A[MxK] @ Wq[NxK]^T) + bias[N].
// Block: 256 thr = 8 waves; block tile 32(M)x256(N); wave tile 16x64 (A reuse x4).
// Register double-buffered over K (prefetch k+8 while computing k) so the
// s_wait_loadcnt before each WMMA group targets loads issued 8 WMMAs earlier.
__global__ __launch_bounds__(256) void gemm_tern(
    const float* __restrict__ A, const float* __restrict__ Wq,
    const float* __restrict__ bias, const float* __restrict__ scale_p,
    float* __restrict__ C, int M, int N, int K) {
  const int lane = threadIdx.x & 31;
  const int wave = threadIdx.x >> 5;
  const int half = lane >> 4;
  const int l16 = lane & 15;
  const int m0 = blockIdx.y * 32 + (wave & 1) * 16;
  const int n0 = blockIdx.x * 256 + (wave >> 1) * 64;

  int arow = m0 + l16;
  if (arow >= M) arow = M - 1;  // clamped load; store is guarded
  const float* Ap = A + (size_t)arow * K + half * 4;
  const float* Bp = Wq + (size_t)(n0 + l16) * K + half * 4;

  v8f c[4] = {};
  v4f ab = *(const v4f*)(Ap);
  v4f bb[4];
#pragma unroll
  for (int j = 0; j < 4; ++j) bb[j] = *(const v4f*)(Bp + (size_t)j * 16 * K);

  for (int kk = 0; kk < K; kk += 8) {
    const int kn = (kk + 8 < K) ? kk + 8 : 0;  // last-iter prefetch is dummy
    v4f an = *(const v4f*)(Ap + kn);
    v4f bn[4];
#pragma unroll
    for (int j = 0; j < 4; ++j)
      bn[j] = *(const v4f*)(Bp + (size_t)j * 16 * K + kn);

    v2f a1 = {ab[0], ab[1]};
    v2f a2 = {ab[2], ab[3]};
#pragma unroll
    for (int j = 0; j < 4; ++j) {
      v2f b1 = {bb[j][0], bb[j][1]};
      c[j] = wmma4(a1, b1, c[j]);
    }
#pragma unroll
    for (int j = 0; j < 4; ++j) {
      v2f b2 = {bb[j][2], bb[j][3]};
      c[j] = wmma4(a2, b2, c[j]);
    }
    ab = an;
#pragma unroll
    for (int j = 0; j < 4; ++j) bb[j] = bn[j];
  }

  const float s = *scale_p;
#pragma unroll
  for (int j = 0; j < 4; ++j) {
    const int n = n0 + j * 16 + l16;
    const float bv = bias[n];
#pragma unroll
    for (int r = 0; r < 8; ++r) {
      const int m = m0 + r + 8 * half;  // C layout: vgpr r -> row r (+8 hi half)
      if (m < M) C[(size_t)m * N + n] = s * c[j][r] + bv;
    }
  }
}

// ------------------------------------------------------------------ layernorm
__global__ __launch_bounds__(256) void ln_kernel(const float* __restrict__ in,
                                                 const float* __restrict__ g,
                                                 const float* __restrict__ bb,
                                                 float* __restrict__ out) {
  const int t = threadIdx.x;
  const float* x = in + (size_t)blockIdx.x * H_;
  __shared__ float red[256];
  float x0 = x[t], x1 = x[t + 256], x2 = x[t + 512];
  red[t] = x0 + x1 + x2;
  __syncthreads();
  for (int o = 128; o > 0; o >>= 1) {
    if (t < o) red[t] += red[t + o];
    __syncthreads();
  }
  const float mu = red[0] / (float)H_;
  __syncthreads();
  float d0 = x0 - mu, d1 = x1 - mu, d2 = x2 - mu;
  red[t] = d0 * d0 + d1 * d1 + d2 * d2;
  __syncthreads();
  for (int o = 128; o > 0; o >>= 1) {
    if (t < o) red[t] += red[t + o];
    __syncthreads();
  }
  const float inv = rsqrtf(red[0] / (float)H_ + 1e-5f);
  float* o = out + (size_t)blockIdx.x * H_;
  o[t] = d0 * inv * g[t] + bb[t];
  o[t + 256] = d1 * inv * g[t + 256] + bb[t + 256];
  o[t + 512] = d2 * inv * g[t + 512] + bb[t + 512];
}

// ------------------------------------------------------- self-attention (flash)
// One wave per (b,h,16-query block). Score tile computed transposed
// (P = K_t @ Q_blk^T) so each softmax column is lane-resident: per-lane
// max/sum + one shfl_xor(16) merges the two half-wave row groups.
// Q fragments (32 floats/lane) are preloaded once and stay in registers.
__global__ __launch_bounds__(256) void self_attn_kernel(
    const float* __restrict__ qkv, const float* __restrict__ tproj,
    const unsigned char* __restrict__ mask, const float* __restrict__ rel,
    float* __restrict__ tself) {
  const int lane = threadIdx.x & 31;
  const int half = lane >> 4;
  const int l16 = lane & 15;
  const int tile = blockIdx.x * 8 + (threadIdx.x >> 5);
  const int qt = tile & 31;
  const int h = (tile >> 5) % NH_;
  const int b = tile / (32 * NH_);

  const int qrow = qt * 16 + l16;  // this lane's query column
  const float* Qp = qkv + (size_t)(b * S_ + qrow) * QKVW + h * HD_ + half * 4;
  v4f qreg[8];
#pragma unroll
  for (int t8 = 0; t8 < 8; ++t8) qreg[t8] = *(const v4f*)(Qp + 8 * t8);

  v8f o[4] = {};  // out^T (d x q), 4 chunks of 16 d-rows
  float runmax = -1e30f, runsum = 0.f;

  for (int kt = 0; kt < S_ / 16; ++kt) {
    const int krow = kt * 16 + l16;
    const float* Kp =
        qkv + (size_t)(b * S_ + krow) * QKVW + H_ + h * HD_ + half * 4;
    v8f sC = {};
#pragma unroll
    for (int t8 = 0; t8 < 8; ++t8) {
      v4f kv4 = *(const v4f*)(Kp + 8 * t8);  // A: rows = keys, b128 per 8 d
      v2f a1 = {kv4[0], kv4[1]};
      v2f a2 = {kv4[2], kv4[3]};
      v2f b1 = {qreg[t8][0], qreg[t8][1]};   // B: cols = queries (same K perm)
      v2f b2 = {qreg[t8][2], qreg[t8][3]};
      sC = wmma4(a1, b1, sC);
      sC = wmma4(a2, b2, sC);
    }
    float l[8], p[8];
    float tmax = -1e30f;
#pragma unroll
    for (int r = 0; r < 8; ++r) {
      const int kg = kt * 16 + r + 8 * half;  // key index of this element
      int rc = qrow - kg;
      rc = rc < -128 ? -128 : (rc > 128 ? 128 : rc);
      float lv = sC[r] * 0.125f + rel[(rc + 128) * NH_ + h];
      lv = mask[b * S_ + kg] ? lv : -1e9f;
      l[r] = lv;
      tmax = fmaxf(tmax, lv);
    }
    tmax = fmaxf(tmax, __shfl_xor(tmax, 16, 32));
    const float nmax = fmaxf(runmax, tmax);
    const float cf = expf(runmax - nmax);
    float tsum = 0.f;
#pragma unroll
    for (int r = 0; r < 8; ++r) {
      p[r] = expf(l[r] - nmax);
      tsum += p[r];
    }
    tsum += __shfl_xor(tsum, 16, 32);
    runsum = runsum * cf + tsum;
    runmax = nmax;
#pragma unroll
    for (int j = 0; j < 4; ++j) o[j] = o[j] * cf;

    // out^T(d,q) += V_t^T(d x 16k) @ P(16k x 16q): rebuild B-fragments of P
    // from the C-layout score registers (cross-half values via shfl_xor 16).
    const float* Vbase =
        qkv + (size_t)(b * S_ + kt * 16) * QKVW + 2 * H_ + h * HD_;
#pragma unroll
    for (int t = 0; t < 4; ++t) {
      const int rA = (t & 1) * 4;  // regs holding low-half k pair
      const int rB = rA + 2;       // regs holding high-half k pair
      const float sA0 = __shfl_xor(p[rA], 16, 32);
      const float sA1 = __shfl_xor(p[rA + 1], 16, 32);
      const float sB0 = __shfl_xor(p[rB], 16, 32);
      const float sB1 = __shfl_xor(p[rB + 1], 16, 32);
      v2f pb;
      if (t < 2) {  // k 0..7 live in low-half lanes
        pb[0] = half ? sB0 : p[rA];
        pb[1] = half ? sB1 : p[rA + 1];
      } else {      // k 8..15 live in high-half lanes
        pb[0] = half ? p[rB] : sA0;
        pb[1] = half ? p[rB + 1] : sA1;
      }
      const int kA = 4 * t + 2 * half;
#pragma unroll
      for (int j = 0; j < 4; ++j) {
        v2f va;  // A = V^T: row m = d, k = key-in-tile (column access -> scalar)
        va[0] = Vbase[(size_t)kA * QKVW + j * 16 + l16];
        va[1] = Vbase[(size_t)(kA + 1) * QKVW + j * 16 + l16];
        o[j] = wmma4(va, pb, o[j]);
      }
    }
  }
  const float inv = 1.f / runsum;
  const size_t orow = (size_t)(b * S_ + qrow) * H_ + h * HD_;
#pragma unroll
  for (int j = 0; j < 4; ++j)
#pragma unroll
    for (int r = 0; r < 8; ++r) {
      const size_t idx = orow + j * 16 + r + 8 * half;  // d index
      tself[idx] = tproj[idx] + o[j][r] * inv;
    }
}

// --------------------------------------------- i2t attention (1 query, 512 keys)
__global__ __launch_bounds__(256) void i2t_attn_kernel(
    const float* __restrict__ q, const float* __restrict__ kv,
    const unsigned char* __restrict__ mask, float* __restrict__ outp) {
  const int b = blockIdx.x / NH_;
  const int h = blockIdx.x % NH_;
  const int t = threadIdx.x;
  __shared__ float qs[HD_];
  __shared__ float pr[S_];
  __shared__ float red[256];
  if (t < HD_) qs[t] = q[b * H_ + h * HD_ + t];
  __syncthreads();
  float lg[2], lmax = -1e30f;
#pragma unroll
  for (int i = 0; i < 2; ++i) {
    const int s = t + i * 256;
    const float* kp = kv + (size_t)(b * S_ + s) * (2 * H_) + h * HD_;
    float acc = 0.f;
    for (int d = 0; d < HD_; ++d) acc += qs[d] * kp[d];
    acc *= 0.125f;
    acc = mask[b * S_ + s] ? acc : -1e9f;
    lg[i] = acc;
    lmax = fmaxf(lmax, acc);
  }
  red[t] = lmax;
  __syncthreads();
  for (int o = 128; o > 0; o >>= 1) {
    if (t < o) red[t] = fmaxf(red[t], red[t + o]);
    __syncthreads();
  }
  const float m = red[0];
  __syncthreads();
  float ls = 0.f;
#pragma unroll
  for (int i = 0; i < 2; ++i) {
    const float p = expf(lg[i] - m);
    pr[t + i * 256] = p;
    ls += p;
  }
  red[t] = ls;
  __syncthreads();
  for (int o = 128; o > 0; o >>= 1) {
    if (t < o) red[t] += red[t + o];
    __syncthreads();
  }
  const float inv = 1.f / red[0];
  if (t < HD_) {
    float acc = 0.f;
    for (int s = 0; s < S_; ++s)
      acc += pr[s] * kv[(size_t)(b * S_ + s) * (2 * H_) + H_ + h * HD_ + t];
    outp[b * H_ + h * HD_ + t] = acc * inv;
  }
}

// ------------------------------------------------------------- elementwise glue
__global__ void vcross_kernel(const float* __restrict__ vp,
                              const float* __restrict__ i2t,
                              const float* __restrict__ alpha,
                              float* __restrict__ out) {
  const int i = blockIdx.x * 256 + threadIdx.x;
  if (i < B_ * H_) out[i] = vp[i] + alpha[0] * i2t[i];
}

// t2i softmax is over a single key => weights == 1; output is broadcast of t2i_v.
__global__ void tcross_kernel(const float* __restrict__ ts,
                              const float* __restrict__ t2ikv,
                              const float* __restrict__ alpha,
                              float* __restrict__ out) {
  const int i = blockIdx.x * 256 + threadIdx.x;
  if (i < B_ * S_ * H_) {
    const int b = i / (S_ * H_);
    const int c = i % H_;
    out[i] = ts[i] + alpha[0] * t2ikv[b * (2 * H_) + H_ + c];
  }
}

// ---------------------------------------------------------------------- driver
static void launch_gemm(const float* A, const float* Wq, const float* bias,
                        const float* sp, float* C, int M, int N, int K,
                        hipStream_t st) {
  dim3 g(N / 256, (M + 31) / 32);
  gemm_tern<<<g, dim3(256), 0, st>>>(A, Wq, bias, sp, C, M, N, K);
}

// workspace layout (float offsets); total ~155 MB
static constexpr size_t W768 = (size_t)H_ * H_;  // 589824
static constexpr size_t OFF_WQ_VP = 64;
static constexpr size_t OFF_WQ_TP = OFF_WQ_VP + W768;
static constexpr size_t OFF_WQ_TQKV = OFF_WQ_TP + W768;
static constexpr size_t OFF_WQ_I2TQ = OFF_WQ_TQKV + 3 * W768;
static constexpr size_t OFF_WQ_I2TKV = OFF_WQ_I2TQ + W768;
static constexpr size_t OFF_WQ_T2IKV = OFF_WQ_I2TKV + 2 * W768;
static constexpr size_t OFF_WQ_VOUT = OFF_WQ_T2IKV + 2 * W768;
static constexpr size_t OFF_WQ_TOUT = OFF_WQ_VOUT + W768;
static constexpr size_t NTOK = (size_t)B_ * S_;
static constexpr size_t OFF_P = OFF_WQ_TOUT + W768;      // text_proj -> text_cross
static constexpr size_t OFF_T = OFF_P + NTOK * H_;       // t_n -> text_self
static constexpr size_t OFF_Q = OFF_T + NTOK * H_;       // qkv -> i2t_kv
static constexpr size_t OFF_VPJ = OFF_Q + NTOK * QKVW;   // vision_proj
static constexpr size_t OFF_VI2TN = OFF_VPJ + B_ * H_;
static constexpr size_t OFF_I2TQ = OFF_VI2TN + B_ * H_;
static constexpr size_t OFF_T2IKV = OFF_I2TQ + B_ * H_;
static constexpr size_t OFF_I2TOUT = OFF_T2IKV + (size_t)B_ * 2 * H_;
static constexpr size_t OFF_VCROSS = OFF_I2TOUT + B_ * H_;

extern "C" void kernel_launch(void* const* d_in, const int* in_sizes, int n_in,
                              void* d_out, int out_size, void* d_ws,
                              size_t ws_size, hipStream_t stream) {
  (void)in_sizes; (void)n_in; (void)out_size; (void)ws_size;
  float* ws = (float*)d_ws;
  float* sc = ws;  // scales[8]
  const float* vis = (const float*)d_in[0];
  const float* txt = (const float*)d_in[1];
  const unsigned char* mask = (const unsigned char*)d_in[2];
  const float* rel = (const float*)d_in[29];

  WArgs wa;
  wa.w[0] = (const float*)d_in[3];  wa.n[0] = (int)W768;        // vp
  wa.w[1] = (const float*)d_in[5];  wa.n[1] = (int)W768;        // tp
  wa.w[2] = (const float*)d_in[7];  wa.n[2] = (int)(3 * W768);  // tqkv
  wa.w[3] = (const float*)d_in[9];  wa.n[3] = (int)W768;        // i2tq
  wa.w[4] = (const float*)d_in[11]; wa.n[4] = (int)(2 * W768);  // i2tkv
  wa.w[5] = (const float*)d_in[15]; wa.n[5] = (int)(2 * W768);  // t2ikv
  wa.w[6] = (const float*)d_in[25]; wa.n[6] = (int)W768;        // vout
  wa.w[7] = (const float*)d_in[27]; wa.n[7] = (int)W768;        // tout
  absmean_kernel<<<8, 256, 0, stream>>>(wa, sc);

  const size_t wqoff[8] = {OFF_WQ_VP,    OFF_WQ_TP,    OFF_WQ_TQKV,
                           OFF_WQ_I2TQ,  OFF_WQ_I2TKV, OFF_WQ_T2IKV,
                           OFF_WQ_VOUT,  OFF_WQ_TOUT};
  for (int i = 0; i < 8; ++i)
    quantize_kernel<<<2048, 256, 0, stream>>>(wa.w[i], ws + wqoff[i], sc + i,
                                              wa.n[i]);

  // projections
  launch_gemm(vis, ws + OFF_WQ_VP, (const float*)d_in[4], sc + 0, ws + OFF_VPJ,
              B_, H_, H_, stream);
  launch_gemm(txt, ws + OFF_WQ_TP, (const float*)d_in[6], sc + 1, ws + OFF_P,
              (int)NTOK, H_, H_, stream);
  ln_kernel<<<(int)NTOK, 256, 0, stream>>>(ws + OFF_P, (const float*)d_in[19],
                                           (const float*)d_in[20], ws + OFF_T);
  ln_kernel<<<B_, 256, 0, stream>>>(ws + OFF_VPJ, (const float*)d_in[21],
                                    (const float*)d_in[22], ws + OFF_VI2TN);
  // qkv + self attention (text_self overwrites t_n region)
  launch_gemm(ws + OFF_T, ws + OFF_WQ_TQKV, (const float*)d_in[8], sc + 2,
              ws + OFF_Q, (int)NTOK, QKVW, H_, stream);
  self_attn_kernel<<<(B_ * NH_ * (S_ / 16)) / 8, 256, 0, stream>>>(
      ws + OFF_Q, ws + OFF_P, mask, rel, ws + OFF_T);
  // cross paths
  launch_gemm(ws + OFF_VI2TN, ws + OFF_WQ_I2TQ, (const float*)d_in[10], sc + 3,
              ws + OFF_I2TQ, B_, H_, H_, stream);
  launch_gemm(ws + OFF_T, ws + OFF_WQ_I2TKV, (const float*)d_in[12], sc + 4,
              ws + OFF_Q, (int)NTOK, 2 * H_, H_, stream);  // reuse qkv region
  launch_gemm(ws + OFF_VPJ, ws + OFF_WQ_T2IKV, (const float*)d_in[16], sc + 5,
              ws + OFF_T2IKV, B_, 2 * H_, H_, stream);
  i2t_attn_kernel<<<B_ * NH_, 256, 0, stream>>>(ws + OFF_I2TQ, ws + OFF_Q, mask,
                                                ws + OFF_I2TOUT);
  vcross_kernel<<<(B_ * H_ + 255) / 256, 256, 0, stream>>>(
      ws + OFF_VPJ, ws + OFF_I2TOUT, (const float*)d_in[30], ws + OFF_VCROSS);
  tcross_kernel<<<((int)(NTOK * H_) + 255) / 256, 256, 0, stream>>>(
      ws + OFF_T, ws + OFF_T2IKV, (const float*)d_in[31], ws + OFF_P);
  // fused outputs
  launch_gemm(ws + OFF_VCROSS, ws + OFF_WQ_VOUT, (const float*)d_in[26], sc + 6,
              (float*)d_out, B_, H_, H_, stream);
  launch_gemm(ws + OFF_P, ws + OFF_WQ_TOUT, (const float*)d_in[28], sc + 7,
              (float*)d_out + (size_t)B_ * H_, (int)NTOK, H_, H_, stream);
}